// TrackCrop_39015482917198
// MI455X (gfx1250) — compile-verified
//
#include <hip/hip_runtime.h>
#include <stdint.h>

// TrackCrop for MI455X (gfx1250): bandwidth-bound mask-centroid + dynamic crop.
// ws layout (int): [0,32) count  [32,64) sum_x  [64,96) sum_y  [96,128) H  [128,160) W

#define IMG_H   512
#define IMG_W   512
#define NCH     3
#define CROP    256
#define IMG_ELEMS   (IMG_H * IMG_W * NCH)   // 786432 floats / image
#define IMG_VEC4    (IMG_ELEMS / 4)         // 196608 float4 / image
#define ROW_F       (CROP * NCH)            // 768 floats per crop row
#define SRC_ROW_F   (IMG_W * NCH)           // 1536 floats per source row
#define ROWS_PER_BLK 4
#define RED_BLKS    32                      // reduce blocks per image

__global__ void tc_zero(int* __restrict__ ws) {
    int t = threadIdx.x;
    if (t < 96) ws[t] = 0;
}

__global__ __launch_bounds__(256)
void tc_reduce(const float4* __restrict__ in, int* __restrict__ ws) {
    const int img = blockIdx.y;
    const float4* p = in + (size_t)img * IMG_VEC4;
    const int t = blockIdx.x * 256 + threadIdx.x;      // 0 .. 8191
    const int stride = RED_BLKS * 256;                 // 8192

    int cnt = 0, sx = 0, sy = 0;
    for (int i = t; i < IMG_VEC4; i += stride) {       // exactly 24 iters
        float4 v = p[i];                               // global_load_b128
        float f0 = v.x, f1 = v.y, f2 = v.z, f3 = v.w;
        int e = i * 4;
        float ff[4] = {f0, f1, f2, f3};
#pragma unroll
        for (int j = 0; j < 4; ++j) {
            int ee  = e + j;
            int pix = ee / 3;                          // magic-mul division
            int x   = pix & (IMG_W - 1);
            int y   = pix >> 9;
            bool nb = (ff[j] != 0.0f);
            cnt += nb ? 1 : 0;
            sx  += nb ? x : 0;
            sy  += nb ? y : 0;
        }
    }

    // wave32 butterfly reduction (CDNA5 is wave32-only)
#pragma unroll
    for (int o = 16; o > 0; o >>= 1) {
        cnt += __shfl_xor(cnt, o, 32);
        sx  += __shfl_xor(sx,  o, 32);
        sy  += __shfl_xor(sy,  o, 32);
    }
    if ((threadIdx.x & 31) == 0) {
        atomicAdd(&ws[img],      cnt);
        atomicAdd(&ws[32 + img], sx);
        atomicAdd(&ws[64 + img], sy);
    }
}

__global__ void tc_params(int* __restrict__ ws) {
    int i = threadIdx.x;
    if (i < 32) {
        float c = (float)ws[i];
        int x = (int)((float)ws[32 + i] / c);
        int y = (int)((float)ws[64 + i] / c);
        int H = min(max(y - CROP / 2, 0), IMG_H - 1 - CROP);  // clamp to 255
        int W = min(max(x - CROP / 2, 0), IMG_W - 1 - CROP);
        ws[96  + i] = H;
        ws[128 + i] = W;
    }
}

// Crop copy: strided 4B-aligned gather -> LDS (async), coalesced 16B scatter
// out of LDS (async). No VGPR data movement; ASYNCcnt tracks completion.
__global__ __launch_bounds__(256)
void tc_crop(const float* __restrict__ in, float* __restrict__ out,
             const int* __restrict__ ws) {
    __shared__ __align__(16) float lds[ROWS_PER_BLK * ROW_F];  // 3072 f = 12KB

    const int img = blockIdx.y;
    const int H = ws[96 + img];
    const int W = ws[128 + img];
    const int r0 = blockIdx.x * ROWS_PER_BLK;

    const float* src = in  + (size_t)img * IMG_ELEMS
                           + ((size_t)(H + r0) * IMG_W + W) * NCH;
    float*       dst = out + (size_t)img * (CROP * ROW_F) + (size_t)r0 * ROW_F;

    // Phase 1: async gather of 4 source rows (768 floats each, rows strided
    // by 1536 floats, start only 4B-aligned) into contiguous LDS.
#pragma unroll
    for (int k = 0; k < (ROWS_PER_BLK * ROW_F) / 256; ++k) {   // 12 iters
        int j = threadIdx.x + k * 256;                          // 0..3071
        int r = j / ROW_F;
        int c = j - r * ROW_F;
        unsigned lv = (unsigned)(uintptr_t)(&lds[j]);
        uint64_t ga = (uint64_t)(uintptr_t)(src + (size_t)r * SRC_ROW_F + c);
        asm volatile("global_load_async_to_lds_b32 %0, %1, off"
                     :: "v"(lv), "v"(ga) : "memory");
    }
    asm volatile("s_wait_asynccnt 0" ::: "memory");  // this wave's LDS writes done
    __syncthreads();                                  // all waves' LDS writes done

    // Phase 2: async 16B-wide scatter LDS -> contiguous global output.
#pragma unroll
    for (int k = 0; k < (ROWS_PER_BLK * ROW_F) / 4 / 256; ++k) {  // 3 iters
        int j = threadIdx.x + k * 256;                            // 0..767
        unsigned lv = (unsigned)(uintptr_t)(&lds[j * 4]);
        uint64_t ga = (uint64_t)(uintptr_t)(dst + (size_t)j * 4);
        asm volatile("global_store_async_from_lds_b128 %0, %1, off"
                     :: "v"(ga), "v"(lv) : "memory");
    }
    asm volatile("s_wait_asynccnt 0" ::: "memory");
}

extern "C" void kernel_launch(void* const* d_in, const int* in_sizes, int n_in,
                              void* d_out, int out_size, void* d_ws, size_t ws_size,
                              hipStream_t stream) {
    const float* in  = (const float*)d_in[0];
    float*       out = (float*)d_out;
    int*         ws  = (int*)d_ws;

    tc_zero  <<<1, 96, 0, stream>>>(ws);
    tc_reduce<<<dim3(RED_BLKS, 32), 256, 0, stream>>>((const float4*)in, ws);
    tc_params<<<1, 32, 0, stream>>>(ws);
    tc_crop  <<<dim3(CROP / ROWS_PER_BLK, 32), 256, 0, stream>>>(in, out, ws);
}